// EmbeddingDropout_68169720922708
// MI455X (gfx1250) — compile-verified
//
#include <hip/hip_runtime.h>
#include <hip/hip_bf16.h>
#include <stdint.h>
#include <stddef.h>

// ---------------------------------------------------------------------------
// EmbeddingDropout forward: out[t, :] = weight[words[t], :] * mask[words[t]]
// words: int32[16384], weight: f32[50257,768], mask: f32[50257], out: f32[16384,768]
// Pure gather -> bandwidth bound (~96MB @ 23.3TB/s ~= 4.1us). No WMMA applies.
// CDNA5 path: async global->LDS B128 copies (ASYNCcnt) double-buffered per wave,
// ds_load_b128 consume, non-temporal global_store_b128 produce.
// ---------------------------------------------------------------------------

typedef float v4f __attribute__((ext_vector_type(4)));
typedef int   v4i __attribute__((vector_size(16)));   // builtin's pointee type

#define AS1 __attribute__((address_space(1)))
#define AS3 __attribute__((address_space(3)))

#if defined(__HIP_DEVICE_COMPILE__) && defined(__gfx1250__) && \
    __has_builtin(__builtin_amdgcn_global_load_async_to_lds_b128) && \
    __has_builtin(__builtin_amdgcn_s_wait_asynccnt)
#define USE_ASYNC 1
#else
#define USE_ASYNC 0
#endif

#define DIMC 768
#define SEGS (DIMC / 4)     // 192 float4 per row
#define TOKS 16             // tokens per tile
#define TPT  16             // threads per token (half-wave)
#define JIT  (SEGS / TPT)   // 12 float4 per thread per tile
#define TILE_V4 (TOKS * SEGS)  // 3072 float4 per buffer (48KB)

__global__ __launch_bounds__(256, 1)
void emb_drop_768(const int* __restrict__ words,
                  const float* __restrict__ weight,
                  const float* __restrict__ mask,
                  float* __restrict__ out,
                  int n_tokens, int n_tiles)
{
#if USE_ASYNC
    __shared__ v4f sm[2 * TILE_V4];   // 96KB double buffer

    const int tid = threadIdx.x;
    const int tl  = tid >> 4;         // token-in-tile 0..15 (half-wave granularity)
    const int s0  = tid & 15;         // seg phase 0..15
    const int stride = gridDim.x;

    int tile = blockIdx.x;
    if (tile >= n_tiles) return;

    AS3 char* lds_base = (AS3 char*)sm;

    // Issue 12 async B128 loads for this wave's tokens of `t` into buffer `buf`;
    // returns the row's dropout mask scalar (prefetched early for latency hiding).
    auto issue = [&](int t, int buf) -> float {
        int token = t * TOKS + tl;
        int ct = token < n_tokens ? token : (n_tokens - 1);
        int row = words[ct];
        float m = mask[row];
        AS1 char* src = (AS1 char*)(void*)(weight) + (size_t)row * (DIMC * 4);
        AS3 char* dst = lds_base + ((size_t)(buf * TOKS + tl) * SEGS) * 16;
        #pragma unroll
        for (int j = 0; j < JIT; ++j) {
            int off = (s0 + j * TPT) * 16;
            __builtin_amdgcn_global_load_async_to_lds_b128(
                (AS1 v4i*)(src + off), (AS3 v4i*)(dst + off), 0, 0);
        }
        return m;
    };

    // Consume this wave's own LDS region (written by its own async loads),
    // scale by mask, stream out with non-temporal B128 stores.
    auto consume = [&](int t, int buf, float m) {
        int token = t * TOKS + tl;
        if (token >= n_tokens) return;
        const v4f* lsrc = (const v4f*)&sm[(buf * TOKS + tl) * SEGS];
        v4f* dst = (v4f*)(out + (size_t)token * DIMC);
        #pragma unroll
        for (int j = 0; j < JIT; ++j) {
            int seg = s0 + j * TPT;
            v4f v = lsrc[seg];
            v *= m;
            __builtin_nontemporal_store(v, dst + seg);
        }
    };

    float m_cur = issue(tile, 0);           // prologue: fill buffer 0
    int buf = 0;
    for (; tile < n_tiles; tile += stride) {
        int ntile = tile + stride;
        float m_next = 0.0f;
        if (ntile < n_tiles) {
            m_next = issue(ntile, buf ^ 1); // keep next stage in flight
            __builtin_amdgcn_s_wait_asynccnt(JIT);  // current stage's 12 done
        } else {
            __builtin_amdgcn_s_wait_asynccnt(0);
        }
        consume(tile, buf, m_cur);
        m_cur = m_next;
        buf ^= 1;
    }
#else
    // Direct-path fallback (also used by host-side parse): B128 loads + NT stores.
    const int tid = threadIdx.x;
    const int tl  = tid >> 4;
    const int s0  = tid & 15;
    for (int tile = blockIdx.x; tile < n_tiles; tile += gridDim.x) {
        int token = tile * TOKS + tl;
        if (token >= n_tokens) continue;
        int row = words[token];
        float m = mask[row];
        const v4f* w = (const v4f*)(weight + (size_t)row * DIMC);
        v4f* dst = (v4f*)(out + (size_t)token * DIMC);
        #pragma unroll
        for (int j = 0; j < JIT; ++j) {
            int seg = s0 + j * TPT;
            v4f v = w[seg];
            v *= m;
            __builtin_nontemporal_store(v, dst + seg);
        }
    }
#endif
}

// Generic fallback for dim != 768 (shape safety only).
__global__ void emb_drop_generic(const int* __restrict__ words,
                                 const float* __restrict__ weight,
                                 const float* __restrict__ mask,
                                 float* __restrict__ out,
                                 long n_tokens, int dim)
{
    int segs = dim >> 2;
    long total = n_tokens * (long)segs;
    for (long i = blockIdx.x * (long)blockDim.x + threadIdx.x; i < total;
         i += (long)gridDim.x * blockDim.x) {
        long token = i / segs;
        int seg = (int)(i - token * segs);
        int row = words[token];
        float m = mask[row];
        const v4f* w = (const v4f*)(weight + (size_t)row * dim);
        v4f v = w[seg];
        v *= m;
        __builtin_nontemporal_store(v, (v4f*)(out + (size_t)token * dim) + seg);
    }
}

extern "C" void kernel_launch(void* const* d_in, const int* in_sizes, int n_in,
                              void* d_out, int out_size, void* d_ws, size_t ws_size,
                              hipStream_t stream)
{
    const int*   words  = (const int*)d_in[0];
    const float* weight = (const float*)d_in[1];
    const float* mask   = (const float*)d_in[2];
    float*       out    = (float*)d_out;

    long n_tokens = in_sizes[0];            // 8*2048 = 16384
    long vocab    = in_sizes[2];            // 50257 (mask is (V,1))
    int  dim      = (int)(in_sizes[1] / vocab);  // 768

    if (dim == DIMC) {
        int n_tiles = (int)((n_tokens + TOKS - 1) / TOKS);   // 1024
        int grid = n_tiles < 512 ? n_tiles : 512;            // 2 tiles/block pipeline
        emb_drop_768<<<grid, 256, 0, stream>>>(words, weight, mask, out,
                                               (int)n_tokens, n_tiles);
    } else {
        long total = n_tokens * (dim >> 2);
        int grid = (int)((total + 255) / 256);
        if (grid > 4096) grid = 4096;
        emb_drop_generic<<<grid, 256, 0, stream>>>(words, weight, mask, out,
                                                   n_tokens, dim);
    }
}